// HyperLatticeBlock_26817775796985
// MI455X (gfx1250) — compile-verified
//
#include <hip/hip_runtime.h>
#include <math.h>

// HyperLatticeBlock: B=2,S=256,D=256,E=64,topk=6 -> grouped-MoE WMMA pipeline.
#define T_TOK 512
#define D_DIM 256
#define E_EXP 64
#define TOPK  6

typedef __attribute__((ext_vector_type(2))) float v2f;
typedef __attribute__((ext_vector_type(8))) float v8f;

__device__ inline v8f wmma4(v2f a, v2f b, v8f c) {
  // V_WMMA_F32_16X16X4_F32: (neg_a, A, neg_b, B, c_mod, C, reuse_a, reuse_b)
  return __builtin_amdgcn_wmma_f32_16x16x4_f32(false, a, false, b, (short)0, c,
                                               false, false);
}

// ---------- setup kernels ----------
__global__ void zero_counts_kernel(int* __restrict__ counts, int* __restrict__ cursor) {
  int i = threadIdx.x;
  if (i < E_EXP) { counts[i] = 0; cursor[i] = 0; }
}

__global__ void transpose_ow_kernel(const float* __restrict__ ow, float* __restrict__ ow_t) {
  int e = blockIdx.x, d = threadIdx.x;           // 256 x 256
  ow_t[d * D_DIM + e] = ow[e * D_DIM + d];
}

__global__ void transpose_gw_kernel(const float* __restrict__ gw, float* __restrict__ gw_t) {
  int e = blockIdx.x, d = threadIdx.x;           // 64 blocks x 256 threads
  gw_t[d * E_EXP + e] = gw[e * D_DIM + d];
}

// ---------- router: logits + top-6 + softmax ----------
__global__ __launch_bounds__(64) void gate_topk_kernel(
    const float* __restrict__ x, const float* __restrict__ gw_t,
    int* __restrict__ indices, float* __restrict__ scores, int* __restrict__ counts) {
  __shared__ float xr[D_DIM];
  __shared__ float lg[E_EXP];
  int t = blockIdx.x, tid = threadIdx.x;
  const float* xrow = x + t * D_DIM;
  for (int i = 0; i < 4; ++i) xr[tid * 4 + i] = xrow[tid * 4 + i];
  __syncthreads();
  float s = 0.f;
  for (int d = 0; d < D_DIM; ++d) s = fmaf(xr[d], gw_t[d * E_EXP + tid], s);
  lg[tid] = s;
  __syncthreads();
  if (tid == 0) {
    float top[TOPK]; int ti[TOPK];
    for (int i = 0; i < TOPK; ++i) {
      float m = -3.0e38f; int mi = 0;
      for (int e = 0; e < E_EXP; ++e) if (lg[e] > m) { m = lg[e]; mi = e; }
      top[i] = m; ti[i] = mi; lg[mi] = -3.0e38f;
    }
    float mx = top[0], den = 0.f, ex[TOPK];
    for (int i = 0; i < TOPK; ++i) { ex[i] = __expf(top[i] - mx); den += ex[i]; }
    float rden = 1.f / den;
    for (int i = 0; i < TOPK; ++i) {
      indices[t * TOPK + i] = ti[i];
      scores[t * TOPK + i]  = ex[i] * rden;
      atomicAdd(&counts[ti[i]], 1);
    }
  }
}

// ---------- exclusive scan of 64 expert counts ----------
__global__ void scan_kernel(const int* __restrict__ counts, int* __restrict__ offsets) {
  if (threadIdx.x == 0) {
    int acc = 0;
    for (int e = 0; e < E_EXP; ++e) { offsets[e] = acc; acc += counts[e]; }
    offsets[E_EXP] = acc;
  }
}

// ---------- scatter (token*6+slot, score) into expert buckets ----------
__global__ __launch_bounds__(256) void scatter_kernel(
    const int* __restrict__ indices, const float* __restrict__ scores,
    const int* __restrict__ offsets, int* __restrict__ cursor,
    int* __restrict__ bucket_dst, float* __restrict__ bucket_score) {
  int idx = blockIdx.x * 256 + threadIdx.x;
  if (idx >= T_TOK * TOPK) return;
  int e = indices[idx];
  int p = offsets[e] + atomicAdd(&cursor[e], 1);
  bucket_dst[p]   = idx;            // row key = t*6 + slot
  bucket_score[p] = scores[idx];
}

// ---------- grouped expert GEMM: xw[t*6+slot][:] = score * (x[t] @ W_e) ----------
__global__ __launch_bounds__(256) void expert_gemm_kernel(
    const float* __restrict__ x, const float* __restrict__ lw,
    const int* __restrict__ offsets, const int* __restrict__ bucket_dst,
    const float* __restrict__ bucket_score, float* __restrict__ xw) {
  int e = blockIdx.x, mt = blockIdx.y;
  int start = offsets[e], end = offsets[e + 1];
  if (mt * 16 >= end - start) return;

  __shared__ float a_s[16 * D_DIM];
  __shared__ int   dst_s[16];
  __shared__ float sc_s[16];
  int tid = threadIdx.x;
  if (tid < 16) {
    int g = start + mt * 16 + tid;
    if (g < end) { dst_s[tid] = bucket_dst[g]; sc_s[tid] = bucket_score[g]; }
    else         { dst_s[tid] = -1;            sc_s[tid] = 0.f; }
  }
  __syncthreads();
  {   // stage A tile: 16 gathered, score-scaled rows of x (zero-padded)
    int r = tid >> 4, c0 = (tid & 15) * 16;
    int dst = dst_s[r];
    float sc = sc_s[r];
    const float* xr = x + (size_t)((dst >= 0 ? dst : 0) / TOPK) * D_DIM;
    for (int i = 0; i < 16; ++i)
      a_s[r * D_DIM + c0 + i] = (dst >= 0) ? xr[c0 + i] * sc : 0.f;
  }
  __syncthreads();

  const float* W = lw + (size_t)e * D_DIM * D_DIM;
  int lane = tid & 31, w = tid >> 5;
  int half = lane >> 4, l15 = lane & 15, koff = half * 2;
  int n0 = (w * 2) * 16, n1 = (w * 2 + 1) * 16;   // wave covers cols [32w, 32w+32)
  v8f c0 = {}; v8f c1 = {};
  for (int k = 0; k < D_DIM; k += 4) {
    v2f a, b0, b1;
    a.x  = a_s[l15 * D_DIM + k + koff];
    a.y  = a_s[l15 * D_DIM + k + koff + 1];
    b0.x = W[(k + koff)     * D_DIM + n0 + l15];
    b0.y = W[(k + koff + 1) * D_DIM + n0 + l15];
    b1.x = W[(k + koff)     * D_DIM + n1 + l15];
    b1.y = W[(k + koff + 1) * D_DIM + n1 + l15];
    c0 = wmma4(a, b0, c0);
    c1 = wmma4(a, b1, c1);
  }
#pragma unroll
  for (int r = 0; r < 8; ++r) {
    int m = r + half * 8;
    int dst = dst_s[m];
    if (dst >= 0) {
      xw[(size_t)dst * D_DIM + n0 + l15] = c0[r];
      xw[(size_t)dst * D_DIM + n1 + l15] = c1[r];
    }
  }
}

// ---------- combine slots + out projection (WMMA) + residual + LayerNorm ----------
__global__ __launch_bounds__(256) void proj_ln_kernel(
    const float* __restrict__ x, const float* __restrict__ xw,
    const float* __restrict__ ow_t, const float* __restrict__ out_b,
    const float* __restrict__ ln_g, const float* __restrict__ ln_b,
    float* __restrict__ out) {
  __shared__ float a_s[16 * D_DIM];
  __shared__ float h_s[16 * D_DIM];
  int tile = blockIdx.x, tid = threadIdx.x;
  {   // A tile: lattice_out rows = sum over 6 slots (scores already folded in)
    int r = tid >> 4, c0 = (tid & 15) * 16;
    int t = tile * 16 + r;
    for (int i = 0; i < 16; ++i) {
      float s = 0.f;
      for (int k = 0; k < TOPK; ++k)
        s += xw[((size_t)t * TOPK + k) * D_DIM + c0 + i];
      a_s[r * D_DIM + c0 + i] = s;
    }
  }
  __syncthreads();
  int lane = tid & 31, w = tid >> 5;
  int half = lane >> 4, l15 = lane & 15, koff = half * 2;
  int n0 = (w * 2) * 16, n1 = (w * 2 + 1) * 16;
  v8f c0 = {}; v8f c1 = {};
  for (int k = 0; k < D_DIM; k += 4) {
    v2f a, b0, b1;
    a.x  = a_s[l15 * D_DIM + k + koff];
    a.y  = a_s[l15 * D_DIM + k + koff + 1];
    b0.x = ow_t[(k + koff)     * D_DIM + n0 + l15];
    b0.y = ow_t[(k + koff + 1) * D_DIM + n0 + l15];
    b1.x = ow_t[(k + koff)     * D_DIM + n1 + l15];
    b1.y = ow_t[(k + koff + 1) * D_DIM + n1 + l15];
    c0 = wmma4(a, b0, c0);
    c1 = wmma4(a, b1, c1);
  }
#pragma unroll
  for (int r = 0; r < 8; ++r) {           // h = x + proj + bias into LDS
    int m = r + half * 8;
    int t = tile * 16 + m;
    int cA = n0 + l15, cB = n1 + l15;
    h_s[m * D_DIM + cA] = c0[r] + out_b[cA] + x[(size_t)t * D_DIM + cA];
    h_s[m * D_DIM + cB] = c1[r] + out_b[cB] + x[(size_t)t * D_DIM + cB];
  }
  __syncthreads();
  for (int mm = 0; mm < 2; ++mm) {        // wave w normalizes tokens 2w, 2w+1
    int m = w * 2 + mm;
    int t = tile * 16 + m;
    float s = 0.f;
    for (int j = 0; j < 8; ++j) s += h_s[m * D_DIM + lane + 32 * j];
    for (int o = 16; o > 0; o >>= 1) s += __shfl_xor(s, o, 32);
    float mean = s * (1.f / 256.f);
    float v = 0.f;
    for (int j = 0; j < 8; ++j) {
      float d = h_s[m * D_DIM + lane + 32 * j] - mean;
      v = fmaf(d, d, v);
    }
    for (int o = 16; o > 0; o >>= 1) v += __shfl_xor(v, o, 32);
    float rinv = rsqrtf(v * (1.f / 256.f) + 1e-5f);
    for (int j = 0; j < 8; ++j) {
      int c = lane + 32 * j;
      out[(size_t)t * D_DIM + c] = (h_s[m * D_DIM + c] - mean) * rinv * ln_g[c] + ln_b[c];
    }
  }
}

extern "C" void kernel_launch(void* const* d_in, const int* in_sizes, int n_in,
                              void* d_out, int out_size, void* d_ws, size_t ws_size,
                              hipStream_t stream) {
  (void)in_sizes; (void)n_in; (void)out_size; (void)ws_size;
  const float* x    = (const float*)d_in[0];   // [2,256,256]
  const float* gw   = (const float*)d_in[1];   // [64,256]
  const float* lw   = (const float*)d_in[2];   // [64,256,256]
  const float* ow   = (const float*)d_in[3];   // [256,256]
  const float* ob   = (const float*)d_in[4];   // [256]
  const float* lg   = (const float*)d_in[5];   // [256]
  const float* lb   = (const float*)d_in[6];   // [256]
  float* out = (float*)d_out;

  char* p = (char*)d_ws;
  auto carve = [&p](size_t bytes) {
    void* r = (void*)p;
    p += (bytes + 255) & ~(size_t)255;
    return r;
  };
  float* ow_t         = (float*)carve((size_t)D_DIM * D_DIM * 4);
  float* gw_t         = (float*)carve((size_t)D_DIM * E_EXP * 4);
  int*   indices      = (int*)  carve((size_t)T_TOK * TOPK * 4);
  float* scores       = (float*)carve((size_t)T_TOK * TOPK * 4);
  int*   counts       = (int*)  carve(E_EXP * 4);
  int*   cursor       = (int*)  carve(E_EXP * 4);
  int*   offsets      = (int*)  carve((E_EXP + 1) * 4);
  int*   bucket_dst   = (int*)  carve((size_t)T_TOK * TOPK * 4);
  float* bucket_score = (float*)carve((size_t)T_TOK * TOPK * 4);
  float* xw           = (float*)carve((size_t)T_TOK * TOPK * D_DIM * 4);  // 3 MB

  zero_counts_kernel<<<1, 64, 0, stream>>>(counts, cursor);
  transpose_ow_kernel<<<D_DIM, D_DIM, 0, stream>>>(ow, ow_t);
  transpose_gw_kernel<<<E_EXP, D_DIM, 0, stream>>>(gw, gw_t);
  gate_topk_kernel<<<T_TOK, 64, 0, stream>>>(x, gw_t, indices, scores, counts);
  scan_kernel<<<1, 32, 0, stream>>>(counts, offsets);
  scatter_kernel<<<(T_TOK * TOPK + 255) / 256, 256, 0, stream>>>(
      indices, scores, offsets, cursor, bucket_dst, bucket_score);
  dim3 g5(E_EXP, 32);  // worst case: one expert takes all 512 rows -> 32 M-tiles
  expert_gemm_kernel<<<g5, 256, 0, stream>>>(x, lw, offsets, bucket_dst,
                                             bucket_score, xw);
  proj_ln_kernel<<<T_TOK / 16, 256, 0, stream>>>(x, xw, ow_t, ob, lg, lb, out);
}